// VanillaMLA_50457275794047
// MI455X (gfx1250) — compile-verified
//
#include <hip/hip_runtime.h>
#include <hip/hip_bf16.h>
#include <math.h>

typedef __attribute__((ext_vector_type(2))) float v2f;
typedef __attribute__((ext_vector_type(8))) float v8f;

#define ATT_SCALE 0.07216878364870323f /* (128+64)^-0.5 */

#if defined(__HIP_DEVICE_COMPILE__) && \
    __has_builtin(__builtin_amdgcn_tensor_load_to_lds)
#define USE_TDM 1
#else
#define USE_TDM 0
#endif

#if USE_TDM
typedef __attribute__((ext_vector_type(4))) unsigned int u32x4;
typedef __attribute__((ext_vector_type(8))) int i32x8;
typedef __attribute__((ext_vector_type(4))) int i32x4;

// 2D tile load via Tensor Data Mover.  Element = 4B.  LDS padding:
// pad_amount(+1) DWORDs inserted after every 2^(padI+1) DWORDs stored.
static __device__ __forceinline__ void tdm_load_2d(unsigned ldsOff,
                                                   const void* g, unsigned tx,
                                                   unsigned ty, unsigned strideX,
                                                   unsigned padI, unsigned padA) {
    unsigned long long ga = (unsigned long long)(size_t)g;
    u32x4 g0;
    g0[0] = 1u;                                                // count=1
    g0[1] = ldsOff;                                            // lds_addr
    g0[2] = (unsigned)ga;                                      // gaddr[31:0]
    g0[3] = (unsigned)((ga >> 32) & 0x01FFFFFFull) | (2u << 30);  // gaddr hi | type=2
    i32x8 g1;
    g1[0] = (int)((2u << 16) | (1u << 20) | (padI << 22) | (padA << 25));
    g1[1] = (int)((tx & 0xFFFFu) << 16);                  // tensor_dim0 lo
    g1[2] = (int)((tx >> 16) | ((ty & 0xFFFFu) << 16));   // dim0 hi | dim1 lo
    g1[3] = (int)((ty >> 16) | (tx << 16));               // dim1 hi | tile_dim0
    g1[4] = (int)ty;                                      // tile_dim1
    g1[5] = (int)strideX;                                 // dim0_stride lo
    g1[6] = 0;
    g1[7] = 0;
    i32x4 z4;
    z4[0] = 0; z4[1] = 0; z4[2] = 0; z4[3] = 0;
    i32x8 z8;
    z8[0] = 0; z8[1] = 0; z8[2] = 0; z8[3] = 0;
    z8[4] = 0; z8[5] = 0; z8[6] = 0; z8[7] = 0;
    __builtin_amdgcn_tensor_load_to_lds(g0, g1, z4, z4, z8, 0);
}

static __device__ __forceinline__ void tensor_wait() {
#if __has_builtin(__builtin_amdgcn_s_wait_tensorcnt)
    __builtin_amdgcn_s_wait_tensorcnt(0);
#else
    asm volatile("s_wait_tensorcnt 0x0" ::: "memory");
#endif
}
#endif

static __device__ __forceinline__ v8f wmma4(v2f a, v2f b, v8f c) {
    // V_WMMA_F32_16X16X4_F32 : D = A(16x4) * B(4x16) + C(16x16), all f32
    return __builtin_amdgcn_wmma_f32_16x16x4_f32(false, a, false, b, (short)0, c,
                                                 false, false);
}

// ---------------------------------------------------------------------------
// Generic tiled WMMA-f32 GEMM:  C[M,N] = A[M,K] @ B[K,N]   (BT: B is N x K)
// block = 128 threads (4 waves), block tile 128x64, wave tile 64x32.
// Double-buffered LDS; A (and B when !BT) staged by the Tensor Data Mover.
// grid = (N/64, M/128, batch)
// ---------------------------------------------------------------------------
template <bool BT>
__global__ __launch_bounds__(128) void gemm_wmma(
    const float* __restrict__ A, long long aB, int lda,
    const float* __restrict__ B, long long bB, int ldb,
    float* __restrict__ C, long long cB, int ldc, int K) {
    __shared__ float lA[2][128 * 36];  // A tile 128x32, row stride 36 (pad)
    __shared__ float lB[2][32 * 68];   // B tile 32x64,  row stride 68 (pad)

    const int tid = threadIdx.x;
    const int lane = tid & 31, w = tid >> 5;
    const int l16 = lane & 15, half = lane >> 4;
    const int m0 = blockIdx.y * 128, n0 = blockIdx.x * 64;
    const int wm = (w >> 1) * 64;  // wave offset in block tile (2x2 waves)
    const int wn = (w & 1) * 32;

    A += (long long)blockIdx.z * aB;
    B += (long long)blockIdx.z * bB;
    C += (long long)blockIdx.z * cB;

    v8f acc[4][2];
#pragma unroll
    for (int i = 0; i < 4; ++i)
#pragma unroll
        for (int j = 0; j < 2; ++j)
#pragma unroll
            for (int v = 0; v < 8; ++v) acc[i][j][v] = 0.0f;

    auto stage = [&](int kc, int b) {
#if USE_TDM
        if (w == 0) {
            tdm_load_2d((unsigned)(size_t)&lA[b][0],
                        A + (size_t)m0 * lda + kc, 32, 128, (unsigned)lda,
                        /*interval 32dw*/ 4, /*pad 4dw*/ 3);
            if (!BT)
                tdm_load_2d((unsigned)(size_t)&lB[b][0],
                            B + (size_t)kc * ldb + n0, 64, 32, (unsigned)ldb,
                            /*interval 64dw*/ 5, /*pad 4dw*/ 3);
        }
        if (BT) {  // manual transposed staging of B (N x K row-major)
#pragma unroll
            for (int it = 0; it < 4; ++it) {
                const int n = it * 16 + (tid >> 3);
                const int k4 = tid & 7;
                float4 v = *reinterpret_cast<const float4*>(
                    B + (size_t)(n0 + n) * ldb + kc + k4 * 4);
                lB[b][(k4 * 4 + 0) * 68 + n] = v.x;
                lB[b][(k4 * 4 + 1) * 68 + n] = v.y;
                lB[b][(k4 * 4 + 2) * 68 + n] = v.z;
                lB[b][(k4 * 4 + 3) * 68 + n] = v.w;
            }
        }
#else
#pragma unroll
        for (int it = 0; it < 8; ++it) {
            const int r = it * 16 + (tid >> 3);
            const int c4 = tid & 7;
            float4 v = *reinterpret_cast<const float4*>(
                A + (size_t)(m0 + r) * lda + kc + c4 * 4);
            float* d = &lA[b][r * 36 + c4 * 4];
            d[0] = v.x; d[1] = v.y; d[2] = v.z; d[3] = v.w;
        }
        if (!BT) {
#pragma unroll
            for (int it = 0; it < 4; ++it) {
                const int r = it * 8 + (tid >> 4);
                const int c4 = tid & 15;
                float4 v = *reinterpret_cast<const float4*>(
                    B + (size_t)(kc + r) * ldb + n0 + c4 * 4);
                float* d = &lB[b][r * 68 + c4 * 4];
                d[0] = v.x; d[1] = v.y; d[2] = v.z; d[3] = v.w;
            }
        } else {
#pragma unroll
            for (int it = 0; it < 4; ++it) {
                const int n = it * 16 + (tid >> 3);
                const int k4 = tid & 7;
                float4 v = *reinterpret_cast<const float4*>(
                    B + (size_t)(n0 + n) * ldb + kc + k4 * 4);
                lB[b][(k4 * 4 + 0) * 68 + n] = v.x;
                lB[b][(k4 * 4 + 1) * 68 + n] = v.y;
                lB[b][(k4 * 4 + 2) * 68 + n] = v.z;
                lB[b][(k4 * 4 + 3) * 68 + n] = v.w;
            }
        }
#endif
    };

    stage(0, 0);
    int buf = 0;
    for (int kc = 0; kc < K; kc += 32) {
#if USE_TDM
        tensor_wait();
#endif
        __syncthreads();
        if (kc + 32 < K) stage(kc + 32, buf ^ 1);  // prefetch next chunk
        // ---- 8 WMMA k-steps over this 32-wide chunk
#pragma unroll
        for (int kk = 0; kk < 32; kk += 4) {
            v2f af[4], bf[2];
#pragma unroll
            for (int mi = 0; mi < 4; ++mi)
                af[mi] = *reinterpret_cast<const v2f*>(
                    &lA[buf][(wm + mi * 16 + l16) * 36 + kk + 2 * half]);
#pragma unroll
            for (int ni = 0; ni < 2; ++ni) {
                const int n = wn + ni * 16 + l16;
                const int k = kk + 2 * half;
                v2f b;
                b.x = lB[buf][k * 68 + n];
                b.y = lB[buf][(k + 1) * 68 + n];
                bf[ni] = b;
            }
#pragma unroll
            for (int mi = 0; mi < 4; ++mi)
#pragma unroll
                for (int ni = 0; ni < 2; ++ni)
                    acc[mi][ni] = wmma4(af[mi], bf[ni], acc[mi][ni]);
        }
        buf ^= 1;
    }
    // ---- store C
#pragma unroll
    for (int mi = 0; mi < 4; ++mi)
#pragma unroll
        for (int ni = 0; ni < 2; ++ni)
#pragma unroll
            for (int v = 0; v < 8; ++v) {
                const int r = m0 + wm + mi * 16 + v + 8 * half;
                const int c = n0 + wn + ni * 16 + l16;
                C[(size_t)r * ldc + c] = acc[mi][ni][v];
            }
}

// ---------------------------------------------------------------------------
// In-place row RMSNorm over `cols` leading columns of rows with stride
// ---------------------------------------------------------------------------
__global__ __launch_bounds__(256) void rmsnorm_kernel(float* X, int stride,
                                                      const float* __restrict__ W,
                                                      int cols) {
    float* x = X + (size_t)blockIdx.x * stride;
    float ss = 0.0f;
    for (int c = threadIdx.x; c < cols; c += 256) {
        const float v = x[c];
        ss += v * v;
    }
    __shared__ float red[256];
    red[threadIdx.x] = ss;
    __syncthreads();
    for (int s = 128; s > 0; s >>= 1) {
        if (threadIdx.x < s) red[threadIdx.x] += red[threadIdx.x + s];
        __syncthreads();
    }
    const float inv = rsqrtf(red[0] / (float)cols + 1e-6f);
    for (int c = threadIdx.x; c < cols; c += 256) x[c] = x[c] * inv * W[c];
}

// ---------------------------------------------------------------------------
// RoPE, interleaved pairs, dim=64.  grid=(seq, heads), block=32 (pair index)
// ---------------------------------------------------------------------------
__global__ __launch_bounds__(32) void rope_kernel(float* X, int rowStride,
                                                  int headStride, int offset) {
    const int s = blockIdx.x, h = blockIdx.y, i = threadIdx.x;  // i in [0,32)
    const float freq = powf(10000.0f, -(float)(2 * i) / 64.0f);
    const float ang = (float)s * freq;
    const float c = cosf(ang), sn = sinf(ang);
    float* p = X + (size_t)s * rowStride + h * headStride + offset + 2 * i;
    const float x0 = p[0], x1 = p[1];
    p[0] = x0 * c - x1 * sn;
    p[1] = x0 * sn + x1 * c;
}

// ---------------------------------------------------------------------------
// Causal flash attention in the 512-d latent space + 64-d rope side channel.
// grid = (2048/128 q-tiles, 32 heads), block = 256 (8 waves, 16 q-rows each).
// Dynamic LDS (~304 KB): the block's whole 128x512 q_lat tile is staged once,
// then each 16-key kv/k_pe tile is streamed in (TDM when available).
// q    : (2048, 32, 192)  row stride 6144   (q_pe at +128 within head)
// qlat : (2048, 32, 512)  row stride 16384  -- read q_lat, write O in place
// kv   : (2048, 576)      cols [0,512)=kv_norm, [512,576)=k_pe
// LDS row strides: q_lat/kv 516 with +2 pad after col 256; k_pe 68; P 18.
// ---------------------------------------------------------------------------
#define LQ_OFF 0
#define LKV_OFF 66048
#define LPE_OFF 74304
#define LP_OFF 75392
#define ATT_LDS_FLOATS 77696

__global__ __launch_bounds__(256) void mla_attn(const float* __restrict__ q,
                                                float* qlat,
                                                const float* __restrict__ kv) {
    extern __shared__ float smem[];
    float* lq = smem + LQ_OFF;    // 128 x 512, stride 516 (+2 mid-row)
    float* lkv = smem + LKV_OFF;  // 16 x 512,  stride 516 (+2 mid-row)
    float* lpe = smem + LPE_OFF;  // 16 x 64,   stride 68

    const int tid = threadIdx.x;
    const int lane = tid & 31, w = tid >> 5;
    const int l16 = lane & 15, half = lane >> 4;
    const int h = blockIdx.y;
    const int qb_blk = blockIdx.x * 128;
    const int qbase = qb_blk + w * 16;
    const int ntiles = blockIdx.x * 8 + 8;  // 16-key tiles this block needs
    float* lPw = smem + LP_OFF + w * 288;   // 16x16 P patch, stride 18

    const float* ape = q + (size_t)(qbase + l16) * 6144 + h * 192 + 128 + 2 * half;
    const float* aql = &lq[(w * 16 + l16) * 516 + 2 * half];
    const float* b0 = &lkv[l16 * 516 + 2 * half];
    const float* p0 = &lpe[l16 * 68 + 2 * half];

    // ---- stage the block's q_lat tile (128 x 512) once
#if USE_TDM
    if (w == 0)
        tdm_load_2d((unsigned)(size_t)lq,
                    qlat + (size_t)qb_blk * 16384 + h * 512, 512, 128, 16384,
                    /*interval 256dw*/ 7, /*pad 2dw*/ 1);
    tensor_wait();
#else
    for (int j = tid; j < 128 * 128; j += 256) {  // float4 granules
        const int r = j >> 7, c4 = j & 127;
        float4 v = *reinterpret_cast<const float4*>(
            qlat + (size_t)(qb_blk + r) * 16384 + h * 512 + c4 * 4);
        float* d = &lq[r * 516 + c4 * 4 + ((c4 >= 64) ? 2 : 0)];
        d[0] = v.x; d[1] = v.y; d[2] = v.z; d[3] = v.w;
    }
#endif
    __syncthreads();

    v8f oacc[32];  // 16 rows x 512 cols of O, f32 accumulators
#pragma unroll
    for (int nf = 0; nf < 32; ++nf)
#pragma unroll
        for (int v = 0; v < 8; ++v) oacc[nf][v] = 0.0f;
    float m_run[8], l_run[8];
#pragma unroll
    for (int v = 0; v < 8; ++v) {
        m_run[v] = -__builtin_inff();
        l_run[v] = 0.0f;
    }

    for (int kt = 0; kt < ntiles; ++kt) {
        const int t0 = kt << 4;
        __syncthreads();  // previous tile fully consumed
        // ---- stage kv (16x512) and k_pe (16x64) tiles
#if USE_TDM
        if (w == 0) {
            tdm_load_2d((unsigned)(size_t)lkv, kv + (size_t)t0 * 576, 512, 16,
                        576, 7, 1);
            tdm_load_2d((unsigned)(size_t)lpe, kv + (size_t)t0 * 576 + 512, 64,
                        16, 576, /*interval 64dw*/ 5, /*pad 4dw*/ 3);
        }
        tensor_wait();
#else
        for (int j = tid; j < 16 * 128; j += 256) {
            const int r = j >> 7, c4 = j & 127;
            float4 v = *reinterpret_cast<const float4*>(
                kv + (size_t)(t0 + r) * 576 + c4 * 4);
            float* d = &lkv[r * 516 + c4 * 4 + ((c4 >= 64) ? 2 : 0)];
            d[0] = v.x; d[1] = v.y; d[2] = v.z; d[3] = v.w;
        }
        {
            const int r = tid >> 4, c4 = tid & 15;
            float4 v = *reinterpret_cast<const float4*>(
                kv + (size_t)(t0 + r) * 576 + 512 + c4 * 4);
            float* d = &lpe[r * 68 + c4 * 4];
            d[0] = v.x; d[1] = v.y; d[2] = v.z; d[3] = v.w;
        }
#endif
        __syncthreads();
        if (t0 > qbase + 15) continue;  // fully-masked tile for this wave

        // ---- S(16q x 16k) = q_lat . kv^T  +  q_pe . k_pe^T
        v8f s0;
#pragma unroll
        for (int v = 0; v < 8; ++v) s0[v] = 0.0f;
#pragma unroll 8
        for (int c = 0; c < 256; c += 4) {
            v2f a = *reinterpret_cast<const v2f*>(aql + c);
            v2f bb = *reinterpret_cast<const v2f*>(b0 + c);
            s0 = wmma4(a, bb, s0);
        }
#pragma unroll 8
        for (int c = 256; c < 512; c += 4) {  // +2 skips mid-row pad
            v2f a = *reinterpret_cast<const v2f*>(aql + c + 2);
            v2f bb = *reinterpret_cast<const v2f*>(b0 + c + 2);
            s0 = wmma4(a, bb, s0);
        }
#pragma unroll
        for (int c = 0; c < 64; c += 4) {
            v2f a = *reinterpret_cast<const v2f*>(ape + c);
            v2f bb = *reinterpret_cast<const v2f*>(p0 + c);
            s0 = wmma4(a, bb, s0);
        }

        // ---- online softmax (row stats per-lane; reduce over 16 lanes)
#pragma unroll
        for (int v = 0; v < 8; ++v) {
            const int qrow = qbase + v + 8 * half;
            float e0 = (t0 + l16 <= qrow) ? s0[v] * ATT_SCALE : -__builtin_inff();
            float mx = e0;
            mx = fmaxf(mx, __shfl_xor(mx, 1, 32));
            mx = fmaxf(mx, __shfl_xor(mx, 2, 32));
            mx = fmaxf(mx, __shfl_xor(mx, 4, 32));
            mx = fmaxf(mx, __shfl_xor(mx, 8, 32));
            const float mnew = fmaxf(m_run[v], mx);
            const float alpha = __expf(m_run[v] - mnew);
            e0 = __expf(e0 - mnew);
            float rs = e0;
            rs += __shfl_xor(rs, 1, 32);
            rs += __shfl_xor(rs, 2, 32);
            rs += __shfl_xor(rs, 4, 32);
            rs += __shfl_xor(rs, 8, 32);
            l_run[v] = l_run[v] * alpha + rs;
            m_run[v] = mnew;
#pragma unroll
            for (int nf = 0; nf < 32; ++nf) oacc[nf][v] *= alpha;
            lPw[(v + 8 * half) * 18 + l16] = e0;  // P transposed via LDS
        }
        asm volatile("s_wait_dscnt 0" ::: "memory");

        // ---- O(16x512) += P(16x16) . KV(16x512)
#pragma unroll
        for (int tt = 0; tt < 16; tt += 4) {
            v2f ap = *reinterpret_cast<const v2f*>(&lPw[l16 * 18 + tt + 2 * half]);
            const int kr = tt + 2 * half;
#pragma unroll
            for (int nf = 0; nf < 32; ++nf) {
                const int cc = nf * 16 + l16 + ((nf >= 16) ? 2 : 0);
                v2f bb;
                bb.x = lkv[kr * 516 + cc];
                bb.y = lkv[(kr + 1) * 516 + cc];
                oacc[nf] = wmma4(ap, bb, oacc[nf]);
            }
        }
    }

    // ---- finalize and write O back over q_lat (rows owned by this wave only)
#pragma unroll
    for (int v = 0; v < 8; ++v) {
        const float inv = 1.0f / l_run[v];
        const size_t rowoff = (size_t)(qbase + v + 8 * half) * 16384 + h * 512;
#pragma unroll
        for (int nf = 0; nf < 32; ++nf)
            qlat[rowoff + nf * 16 + l16] = oacc[nf][v] * inv;
    }
}

// ---------------------------------------------------------------------------
extern "C" void kernel_launch(void* const* d_in, const int* in_sizes, int n_in,
                              void* d_out, int out_size, void* d_ws,
                              size_t ws_size, hipStream_t stream) {
    (void)in_sizes; (void)n_in; (void)out_size; (void)ws_size;
    const float* x       = (const float*)d_in[0];  // (2048,4096)
    const float* wq_a    = (const float*)d_in[1];  // (4096,1536)
    const float* q_norm  = (const float*)d_in[2];  // (1536,)
    const float* wq_b    = (const float*)d_in[3];  // (1536,6144)
    const float* wkv_a   = (const float*)d_in[4];  // (4096,576)
    const float* kv_norm = (const float*)d_in[5];  // (512,)
    const float* wkv_b   = (const float*)d_in[6];  // (32,256,512)
    const float* wo      = (const float*)d_in[7];  // (4096,4096)
    float* out = (float*)d_out;                    // (2048,4096)
    float* ws = (float*)d_ws;

    float* qa    = ws;             // 2048x1536
    float* qbuf  = ws + 3145728;   // 2048x6144 (q: nope|pe per head)
    float* kvf   = ws + 15728640;  // 2048x576  (kv_norm | k_pe)
    float* qlat  = ws + 16908288;  // 2048x32x512 (q_lat, then O in place)
    float* ohead = ws;             // 2048x4096, reuses dead qa/q region

    const dim3 blk(128);

    // 1) qa = x @ wq_a                      M=2048 N=1536 K=4096
    gemm_wmma<false><<<dim3(24, 16, 1), blk, 0, stream>>>(
        x, 0LL, 4096, wq_a, 0LL, 1536, qa, 0LL, 1536, 4096);
    // 2) rmsnorm(qa)
    rmsnorm_kernel<<<2048, 256, 0, stream>>>(qa, 1536, q_norm, 1536);
    // 3) q = qa @ wq_b                      M=2048 N=6144 K=1536
    gemm_wmma<false><<<dim3(96, 16, 1), blk, 0, stream>>>(
        qa, 0LL, 1536, wq_b, 0LL, 6144, qbuf, 0LL, 6144, 1536);
    // 4) kvf = x @ wkv_a                    M=2048 N=576 K=4096
    gemm_wmma<false><<<dim3(9, 16, 1), blk, 0, stream>>>(
        x, 0LL, 4096, wkv_a, 0LL, 576, kvf, 0LL, 576, 4096);
    // 5) rmsnorm(kv part)
    rmsnorm_kernel<<<2048, 256, 0, stream>>>(kvf, 576, kv_norm, 512);
    // 6) rope(q_pe) and rope(k_pe)
    rope_kernel<<<dim3(2048, 32), 32, 0, stream>>>(qbuf, 6144, 192, 128);
    rope_kernel<<<dim3(2048, 1), 32, 0, stream>>>(kvf, 576, 0, 512);
    // 7) q_lat[h] = q_nope[h] @ wkv_b[h,:128,:]   batched, M=2048 N=512 K=128
    gemm_wmma<false><<<dim3(8, 16, 32), blk, 0, stream>>>(
        qbuf, 192LL, 6144, wkv_b, 131072LL, 512, qlat, 512LL, 16384, 128);
    // 8) attention (O written over q_lat); ~304 KB dynamic LDS per block
    mla_attn<<<dim3(16, 32), 256, ATT_LDS_FLOATS * 4, stream>>>(qbuf, qlat,
                                                                kvf);
    // 9) ohead[h] = O[h] @ wkv_b[h,128:,:]^T      batched, M=2048 N=128 K=512
    gemm_wmma<true><<<dim3(2, 16, 32), blk, 0, stream>>>(
        qlat, 512LL, 16384, wkv_b + 128 * 512, 131072LL, 512, ohead, 128LL,
        4096, 512);
    // 10) out = ohead @ wo                  M=2048 N=4096 K=4096
    gemm_wmma<false><<<dim3(64, 16, 1), blk, 0, stream>>>(
        ohead, 0LL, 4096, wo, 0LL, 4096, out, 0LL, 4096, 4096);
}